// MambaBlock_87574383165803
// MI455X (gfx1250) — compile-verified
//
#include <hip/hip_runtime.h>
#include <hip/hip_bf16.h>

// ---------------------------------------------------------------------------
// Mamba block forward on gfx1250 (MI455X, wave32).
//  K1: xz = x @ x_proj   -> u = silu(lo), sz = silu(hi)   [LDS-staged WMMA f32]
//  K2: u @ {dt,B,C}_proj -> dt (softplus+bias), Bm, Cm    [WMMA f32 16x16x4]
//  K3: selective scan (sequential in L), fused skip+gate, yg overwrites u
//  K4: out = yg @ out_proj                                 [LDS-staged WMMA f32]
// ---------------------------------------------------------------------------

typedef __attribute__((ext_vector_type(2))) float v2f;
typedef __attribute__((ext_vector_type(8))) float v8f;

#define BATCH   4
#define SEQLEN  4096
#define DIM     256
#define DINNER  512
#define DSTATE  16
#define MROWS   (BATCH * SEQLEN)   // 16384 token rows

#define BK   64        // K-chunk staged in LDS
#define LDA  68        // padded LDS row stride (272B: 16B-aligned, bank stride 4)

__device__ __forceinline__ float silu_f(float v) {
    return v / (1.0f + __expf(-v));
}

// ---------------------------------------------------------------------------
// Generic LDS-staged 128x64 block GEMM body:
//   C[128 x 64] += A[128 x K] * W[K x ldw] (cols nbase..nbase+63)
// 8 waves/block, each wave one 16x64 strip (4 accumulators).
// A tile: sA[m*LDA + k]  (row-major, K-contiguous)
// B tile: sBt[n*LDA + k] (transposed,  K-contiguous) -> v2f ds_load_b64 frags
// ---------------------------------------------------------------------------

// ------------------------------- K1 ----------------------------------------
// x(16384x256) @ x_proj(256x1024). grid = 128 rowBlks * 16 colBlks = 2048.
__global__ void k_gemm_xz(const float* __restrict__ x,
                          const float* __restrict__ xproj,
                          float* __restrict__ u,
                          float* __restrict__ sz) {
    __shared__ float sA [128 * LDA];   // 34.0 KB
    __shared__ float sBt[ 64 * LDA];   // 17.0 KB
    const int tid  = threadIdx.x;
    const int lane = tid & 31, wv = tid >> 5;
    const int l16  = lane & 15, half = lane >> 4;
    const int rbase = (blockIdx.x >> 4) * 128;
    const int nbase = (blockIdx.x & 15) * 64;

    v8f acc[4] = {};
    for (int k0 = 0; k0 < DIM; k0 += BK) {
        // stage A: 128 rows x 64 k, coalesced b128 loads
        for (int i = tid; i < 128 * 16; i += 256) {
            const int r = i >> 4, c4 = (i & 15) << 2;
            *(float4*)&sA[r * LDA + c4] =
                *(const float4*)&x[(size_t)(rbase + r) * DIM + k0 + c4];
        }
        // stage B^T: 64 k x 64 n -> sBt[n*LDA + k]
        for (int i = tid; i < 64 * 16; i += 256) {
            const int r = i >> 4, c4 = (i & 15) << 2;   // r: k-row, c4: n start
            const float4 v = *(const float4*)&xproj[(size_t)(k0 + r) * 1024 + nbase + c4];
            sBt[(c4 + 0) * LDA + r] = v.x;
            sBt[(c4 + 1) * LDA + r] = v.y;
            sBt[(c4 + 2) * LDA + r] = v.z;
            sBt[(c4 + 3) * LDA + r] = v.w;
        }
        __syncthreads();
        for (int k = 0; k < BK; k += 4) {
            const int kk = k + 2 * half;
            v2f a = *(const v2f*)&sA[(wv * 16 + l16) * LDA + kk];
#pragma unroll
            for (int j = 0; j < 4; ++j) {
                v2f b = *(const v2f*)&sBt[(j * 16 + l16) * LDA + kk];
                acc[j] = __builtin_amdgcn_wmma_f32_16x16x4_f32(
                    false, a, false, b, (short)0, acc[j], false, false);
            }
        }
        __syncthreads();
    }
#pragma unroll
    for (int r = 0; r < 8; ++r) {
        const int row = rbase + wv * 16 + r + 8 * half;
#pragma unroll
        for (int j = 0; j < 4; ++j) {
            const int col = nbase + j * 16 + l16;          // 0..1023
            const float v = silu_f(acc[j][r]);
            if (col < DINNER) u [(size_t)row * DINNER + col]            = v;
            else              sz[(size_t)row * DINNER + (col - DINNER)] = v;
        }
    }
}

// ------------------------------- K2 ----------------------------------------
// u(16384x512) @ dt/B/C_proj(512x16 each). Each wave: 16 rows x 48 cols.
// Weights are 32KB each -> WGP$/L2 resident; direct loads are fine here.
// grid = 128 blocks * 8 waves = 1024 row tiles.
__global__ void k_gemm_dtbc(const float* __restrict__ u,
                            const float* __restrict__ dtw,
                            const float* __restrict__ bw,
                            const float* __restrict__ cw,
                            const float* __restrict__ bias,
                            float* __restrict__ dt,
                            float* __restrict__ Bm,
                            float* __restrict__ Cm) {
    const int lane = threadIdx.x & 31;
    const int wv   = threadIdx.x >> 5;
    const int rbase = (blockIdx.x * 8 + wv) * 16;
    const int l16  = lane & 15;
    const int half = lane >> 4;

    v8f ad = {}, ab = {}, ac = {};
    const float* ur = u + (size_t)(rbase + l16) * DINNER;
    for (int k = 0; k < DINNER; k += 4) {
        const int kk = k + 2 * half;
        v2f a = *(const v2f*)(ur + kk);
        v2f bd = { dtw[kk * 16 + l16], dtw[(kk + 1) * 16 + l16] };
        v2f bb = { bw [kk * 16 + l16], bw [(kk + 1) * 16 + l16] };
        v2f bc = { cw [kk * 16 + l16], cw [(kk + 1) * 16 + l16] };
        ad = __builtin_amdgcn_wmma_f32_16x16x4_f32(false, a, false, bd, (short)0, ad, false, false);
        ab = __builtin_amdgcn_wmma_f32_16x16x4_f32(false, a, false, bb, (short)0, ab, false, false);
        ac = __builtin_amdgcn_wmma_f32_16x16x4_f32(false, a, false, bc, (short)0, ac, false, false);
    }
    const float bia = bias[l16];
#pragma unroll
    for (int r = 0; r < 8; ++r) {
        const int row = rbase + r + 8 * half;
        const size_t idx = (size_t)row * DSTATE + l16;
        const float t = ad[r] + bia;
        dt[idx] = (t > 20.0f) ? t : log1pf(__expf(t));   // softplus
        Bm[idx] = ab[r];
        Cm[idx] = ac[r];
    }
}

// ------------------------------- K3 ----------------------------------------
// Selective scan. 8 blocks x 256 threads: block -> (batch = blk>>1,
// channel range = (blk&1)*256). One lane per (b,d), h[16] in VGPRs.
// dt/B/C staged 64 timesteps at a time into LDS (coalesced load, broadcast
// read). Fused: yg = (y + u*D) * silu(z), written in place over u.
#define CH 64
__global__ void k_scan(const float* __restrict__ dt,
                       const float* __restrict__ Bm,
                       const float* __restrict__ Cm,
                       const float* __restrict__ Alog,
                       const float* __restrict__ Dv,
                       const float* __restrict__ sz,
                       float* __restrict__ u /* in: u, out: y_gated */) {
    __shared__ float sDt[CH * DSTATE];
    __shared__ float sB [CH * DSTATE];
    __shared__ float sC [CH * DSTATE];

    const int b = blockIdx.x >> 1;
    const int d = ((blockIdx.x & 1) << 8) + threadIdx.x;

    float A[DSTATE], h[DSTATE];
#pragma unroll
    for (int s = 0; s < DSTATE; ++s) {
        A[s] = -__expf(Alog[d * DSTATE + s]);
        h[s] = 0.0f;
    }
    const float Dd = Dv[d];

    for (int t0 = 0; t0 < SEQLEN; t0 += CH) {
        const size_t base = (size_t)(b * SEQLEN + t0) * DSTATE;
        for (int i = threadIdx.x; i < CH * DSTATE; i += 256) {
            sDt[i] = dt[base + i];
            sB [i] = Bm[base + i];
            sC [i] = Cm[base + i];
        }
        if (t0 + CH < SEQLEN && threadIdx.x < 32) {
            const size_t nb = base + CH * DSTATE + (size_t)threadIdx.x * 32;
            __builtin_prefetch(dt + nb, 0, 1);   // global_prefetch_b8
            __builtin_prefetch(Bm + nb, 0, 1);
            __builtin_prefetch(Cm + nb, 0, 1);
        }
        __syncthreads();

        for (int tt = 0; tt < CH; ++tt) {
            const size_t row = (size_t)(b * SEQLEN + t0 + tt);
            const float uu = u[row * DINNER + d];
            float y = 0.0f;
#pragma unroll
            for (int s = 0; s < DSTATE; ++s) {
                const float dts = sDt[tt * DSTATE + s];
                const float dA  = __expf(dts * A[s]);
                h[s] = dA * h[s] + dts * sB[tt * DSTATE + s] * uu;
                y += h[s] * sC[tt * DSTATE + s];
            }
            u[row * DINNER + d] = (y + uu * Dd) * sz[row * DINNER + d];
        }
        __syncthreads();
    }
}

// ------------------------------- K4 ----------------------------------------
// yg(16384x512) @ out_proj(512x256) -> out(16384x256).
// grid = 128 rowBlks * 4 colBlks = 512 blocks. Same LDS-staged scheme as K1.
__global__ void k_gemm_out(const float* __restrict__ yg,
                           const float* __restrict__ oproj,
                           float* __restrict__ out) {
    __shared__ float sA [128 * LDA];
    __shared__ float sBt[ 64 * LDA];
    const int tid  = threadIdx.x;
    const int lane = tid & 31, wv = tid >> 5;
    const int l16  = lane & 15, half = lane >> 4;
    const int rbase = (blockIdx.x >> 2) * 128;
    const int nbase = (blockIdx.x & 3) * 64;

    v8f acc[4] = {};
    for (int k0 = 0; k0 < DINNER; k0 += BK) {
        for (int i = tid; i < 128 * 16; i += 256) {
            const int r = i >> 4, c4 = (i & 15) << 2;
            *(float4*)&sA[r * LDA + c4] =
                *(const float4*)&yg[(size_t)(rbase + r) * DINNER + k0 + c4];
        }
        for (int i = tid; i < 64 * 16; i += 256) {
            const int r = i >> 4, c4 = (i & 15) << 2;
            const float4 v = *(const float4*)&oproj[(size_t)(k0 + r) * DIM + nbase + c4];
            sBt[(c4 + 0) * LDA + r] = v.x;
            sBt[(c4 + 1) * LDA + r] = v.y;
            sBt[(c4 + 2) * LDA + r] = v.z;
            sBt[(c4 + 3) * LDA + r] = v.w;
        }
        __syncthreads();
        for (int k = 0; k < BK; k += 4) {
            const int kk = k + 2 * half;
            v2f a = *(const v2f*)&sA[(wv * 16 + l16) * LDA + kk];
#pragma unroll
            for (int j = 0; j < 4; ++j) {
                v2f b = *(const v2f*)&sBt[(j * 16 + l16) * LDA + kk];
                acc[j] = __builtin_amdgcn_wmma_f32_16x16x4_f32(
                    false, a, false, b, (short)0, acc[j], false, false);
            }
        }
        __syncthreads();
    }
#pragma unroll
    for (int r = 0; r < 8; ++r) {
        const int row = rbase + wv * 16 + r + 8 * half;
#pragma unroll
        for (int j = 0; j < 4; ++j) {
            out[(size_t)row * DIM + nbase + j * 16 + l16] = acc[j][r];
        }
    }
}

// ---------------------------------------------------------------------------
extern "C" void kernel_launch(void* const* d_in, const int* in_sizes, int n_in,
                              void* d_out, int out_size, void* d_ws, size_t ws_size,
                              hipStream_t stream) {
    const float* x     = (const float*)d_in[0];   // (4,4096,256)
    const float* xproj = (const float*)d_in[1];   // (256,1024)
    const float* dtw   = (const float*)d_in[2];   // (512,16)
    const float* Alog  = (const float*)d_in[3];   // (512,16)
    const float* bw    = (const float*)d_in[4];   // (512,16)
    const float* cw    = (const float*)d_in[5];   // (512,16)
    const float* Dv    = (const float*)d_in[6];   // (512,)
    const float* oproj = (const float*)d_in[7];   // (512,256)
    const float* bias  = (const float*)d_in[8];   // (16,)
    float* out = (float*)d_out;

    // workspace layout (floats): u | sz | dt | Bm | Cm   (~67 MB total)
    float* u  = (float*)d_ws;                         // 16384*512, becomes yg
    float* sz = u  + (size_t)MROWS * DINNER;          // 16384*512
    float* dt = sz + (size_t)MROWS * DINNER;          // 16384*16
    float* Bm = dt + (size_t)MROWS * DSTATE;
    float* Cm = Bm + (size_t)MROWS * DSTATE;

    k_gemm_xz  <<<2048, 256, 0, stream>>>(x, xproj, u, sz);
    k_gemm_dtbc<<< 128, 256, 0, stream>>>(u, dtw, bw, cw, bias, dt, Bm, Cm);
    k_scan     <<<   8, 256, 0, stream>>>(dt, Bm, Cm, Alog, Dv, sz, u);
    k_gemm_out <<< 512, 256, 0, stream>>>(u, oproj, out);
}